// GraphSAGEEncoder_13142599925969
// MI455X (gfx1250) — compile-verified
//
#include <hip/hip_runtime.h>
#include <hip/hip_bf16.h>
#include <stdint.h>

typedef float v2f __attribute__((ext_vector_type(2)));
typedef float v8f __attribute__((ext_vector_type(8)));

// ---------------------------------------------------------------- utilities
__global__ void zero_kernel(float* __restrict__ p, long long n) {
    long long i = (long long)blockIdx.x * blockDim.x + threadIdx.x;
    long long stride = (long long)gridDim.x * blockDim.x;
    for (; i < n; i += stride) p[i] = 0.0f;
}

__global__ void degree_kernel(const long long* __restrict__ dst,
                              float* __restrict__ deg, long long E) {
    long long e = (long long)blockIdx.x * blockDim.x + threadIdx.x;
    if (e < E) atomicAdd(&deg[dst[e]], 1.0f);
}

__global__ void invdeg_kernel(float* __restrict__ deg, int N) {
    int i = blockIdx.x * blockDim.x + threadIdx.x;
    if (i < N) deg[i] = 1.0f / fmaxf(deg[i], 1.0f);
}

// --------------------------------------------------- edge-parallel scatter-add
// 16 lanes per edge, each lane moves a float4 of the 64-float feature row.
// x / agg rows are L2-resident (25.6MB each vs 192MB L2), so the atomics
// resolve at L2 atomic units.
__global__ void scatter_kernel(const float* __restrict__ h,
                               const long long* __restrict__ src,
                               const long long* __restrict__ dst,
                               float* __restrict__ agg, long long E) {
    long long t = (long long)blockIdx.x * blockDim.x + threadIdx.x;
    long long e = t >> 4;
    int f = (int)(t & 15) * 4;
    if (e >= E) return;
    long long s = src[e];
    long long d = dst[e];
    const float4 v = *(const float4*)(h + s * 64 + f);
    float* p = agg + d * 64 + f;
    atomicAdd(p + 0, v.x);
    atomicAdd(p + 1, v.y);
    atomicAdd(p + 2, v.z);
    atomicAdd(p + 3, v.w);
}

// --------------------------------------------------------- fused node kernel
// out[n,:] = act( (agg[n,:]*inv[n]) @ Wl^T + b + h[n,:] @ Wr^T )
// One wave per 16-node tile, V_WMMA_F32_16X16X4_F32, K looped in steps of 4.
//
// A layout (16x4 f32, ISA 7.12.2): lanes 0-15 hold rows M=0..15 with
// VGPR0=K0,VGPR1=K1; lanes 16-31 hold the same rows with K2,K3.
// B layout (4x16) mirrored: lane n<16 -> B[k0..k0+1, n], lane n+16 -> B[k0+2..k0+3, n].
// B[k][o] = W[o][k], so each lane reads W[o*IN + k0 + koff .. +1].
// C/D: VGPR j, lane l<16 -> D[M=j, N=l]; lane l+16 -> D[M=j+8, N=l-16].
template <int IN, int OUT, bool RELU>
__global__ void sage_node_kernel(const float* __restrict__ h,
                                 const float* __restrict__ agg,
                                 const float* __restrict__ inv,
                                 const float* __restrict__ Wl,
                                 const float* __restrict__ Wr,
                                 const float* __restrict__ bias,
                                 float* __restrict__ out, int N) {
    constexpr int CT = OUT / 16;
    const int wave = (int)((blockIdx.x * blockDim.x + threadIdx.x) >> 5);
    const int lane = threadIdx.x & 31;
    const int row0 = wave * 16;
    if (row0 >= N) return;                 // wave-uniform exit

    const int lrow = lane & 15;            // row within tile (A) / column o (B,D)
    const int koff = (lane >> 4) * 2;      // K sub-offset per half-wave
    int node = row0 + lrow;
    int nodec = node < N ? node : N - 1;   // clamp, keep EXEC all-ones for WMMA
    const float invd = inv[nodec];

    v8f acc[CT];
#pragma unroll
    for (int ct = 0; ct < CT; ++ct) {
        float bv = bias[ct * 16 + lrow];
        v8f t;
#pragma unroll
        for (int j = 0; j < 8; ++j) t[j] = bv;
        acc[ct] = t;
    }

    const float* hrow = h + (long long)nodec * IN;
    const float* arow = agg + (long long)nodec * IN;

#pragma unroll
    for (int k0 = 0; k0 < IN; k0 += 4) {
        v2f a_h, a_m;
        a_h[0] = hrow[k0 + koff + 0];
        a_h[1] = hrow[k0 + koff + 1];
        a_m[0] = arow[k0 + koff + 0] * invd;
        a_m[1] = arow[k0 + koff + 1] * invd;
#pragma unroll
        for (int ct = 0; ct < CT; ++ct) {
            const int o = ct * 16 + lrow;
            const float* pl = Wl + o * IN + k0 + koff;
            const float* pr = Wr + o * IN + k0 + koff;
            v2f bl, br;
            bl[0] = pl[0]; bl[1] = pl[1];
            br[0] = pr[0]; br[1] = pr[1];
            acc[ct] = __builtin_amdgcn_wmma_f32_16x16x4_f32(
                false, a_m, false, bl, (short)0, acc[ct], false, false);
            acc[ct] = __builtin_amdgcn_wmma_f32_16x16x4_f32(
                false, a_h, false, br, (short)0, acc[ct], false, false);
        }
    }

    const int mbase = (lane >> 4) * 8;
#pragma unroll
    for (int ct = 0; ct < CT; ++ct) {
        const int o = ct * 16 + lrow;
#pragma unroll
        for (int j = 0; j < 8; ++j) {
            int nd = row0 + mbase + j;
            if (nd < N) {
                float v = acc[ct][j];
                if (RELU) v = fmaxf(v, 0.0f);
                out[(long long)nd * OUT + o] = v;
            }
        }
    }
}

// -------------------------------------------------------------------- launch
extern "C" void kernel_launch(void* const* d_in, const int* in_sizes, int n_in,
                              void* d_out, int out_size, void* d_ws, size_t ws_size,
                              hipStream_t stream) {
    const float*     x    = (const float*)d_in[0];
    const long long* ei   = (const long long*)d_in[1];
    // d_in[2] = edge_weight, intentionally ignored (GraphSAGE ignores it)
    const float* Wl0 = (const float*)d_in[3];
    const float* Wr0 = (const float*)d_in[4];
    const float* b0  = (const float*)d_in[5];
    const float* Wl1 = (const float*)d_in[6];
    const float* Wr1 = (const float*)d_in[7];
    const float* b1  = (const float*)d_in[8];
    const float* Wl2 = (const float*)d_in[9];
    const float* Wr2 = (const float*)d_in[10];
    const float* b2  = (const float*)d_in[11];

    const int       N = in_sizes[0] / 64;
    const long long E = (long long)in_sizes[1] / 2;
    const long long* src = ei;
    const long long* dst = ei + E;

    // workspace layout
    char*  ws  = (char*)d_ws;
    size_t off = 0;
    auto alloc = [&](size_t bytes) {
        void* p = ws + off;
        off += (bytes + 255) & ~(size_t)255;
        return p;
    };
    float* inv = (float*)alloc((size_t)N * 4);
    float* agg = (float*)alloc((size_t)N * 64 * 4);
    float* h1  = (float*)alloc((size_t)N * 64 * 4);
    float* h2  = (float*)alloc((size_t)N * 64 * 4);
    float* out = (float*)d_out;

    const int ZB   = 2048;                               // zero-fill blocks
    const int eb   = (int)((E + 255) / 256);             // 1 thread / edge
    const int sb   = (int)((E * 16 + 255) / 256);        // 16 threads / edge
    const int nb   = (N + 255) / 256;
    const int tiles = (N + 15) / 16;
    const int wb   = (tiles + 7) / 8;                    // 8 waves / 256-block

    // shared inverse degree (same graph for all layers)
    zero_kernel<<<ZB, 256, 0, stream>>>(inv, N);
    degree_kernel<<<eb, 256, 0, stream>>>(dst, inv, E);
    invdeg_kernel<<<nb, 256, 0, stream>>>(inv, N);

    // layer 0: x -> h1 (ReLU)
    zero_kernel<<<ZB, 256, 0, stream>>>(agg, (long long)N * 64);
    scatter_kernel<<<sb, 256, 0, stream>>>(x, src, dst, agg, E);
    sage_node_kernel<64, 64, true><<<wb, 256, 0, stream>>>(x, agg, inv, Wl0, Wr0, b0, h1, N);

    // layer 1: h1 -> h2 (ReLU)
    zero_kernel<<<ZB, 256, 0, stream>>>(agg, (long long)N * 64);
    scatter_kernel<<<sb, 256, 0, stream>>>(h1, src, dst, agg, E);
    sage_node_kernel<64, 64, true><<<wb, 256, 0, stream>>>(h1, agg, inv, Wl1, Wr1, b1, h2, N);

    // layer 2: h2 -> out (no ReLU, out_dim 32)
    zero_kernel<<<ZB, 256, 0, stream>>>(agg, (long long)N * 64);
    scatter_kernel<<<sb, 256, 0, stream>>>(h2, src, dst, agg, E);
    sage_node_kernel<64, 32, false><<<wb, 256, 0, stream>>>(h2, agg, inv, Wl2, Wr2, b2, out, N);
}